// GCNLayer_14053132992804
// MI455X (gfx1250) — compile-verified
//
#include <hip/hip_runtime.h>
#include <hip/hip_bf16.h>

typedef float v2f __attribute__((ext_vector_type(2)));
typedef float v8f __attribute__((ext_vector_type(8)));

#define D 128          // D_IN == D_OUT == 128
#define SA_STRIDE 132  // padded LDS row stride (floats) -> conflict-free col reads

// ---------------------------------------------------------------------------
// Kernel 1: zero the accumulator (d_out doubles as agg buffer)
// ---------------------------------------------------------------------------
__global__ void gcn_zero_kernel(float4* __restrict__ p, long long n4) {
    long long i = (long long)blockIdx.x * blockDim.x + threadIdx.x;
    if (i < n4) p[i] = make_float4(0.f, 0.f, 0.f, 0.f);
}

// ---------------------------------------------------------------------------
// Kernel 2: x = h @ W  (fp32 WMMA 16x16x4)
// Block = 256 threads = 8 waves. Block handles 16 rows of h; wave w handles
// output columns [16w, 16w+16). K loop: 128/4 = 32 WMMA issues per wave.
// ---------------------------------------------------------------------------
__global__ __launch_bounds__(256)
void gcn_gemm_wmma_kernel(const float* __restrict__ h,
                          const float* __restrict__ W,
                          float* __restrict__ x,
                          int n_nodes) {
    __shared__ float sA[16 * SA_STRIDE];

    const int row_base = blockIdx.x * 16;
    const int t = threadIdx.x;
    const bool full_tile = (row_base + 16 <= n_nodes);  // block-uniform

    // Stage 16x128 tile of h into LDS (512 float4 loads over 256 threads).
    if (full_tile) {
        for (int i = t; i < 16 * 32; i += 256) {
            const int r  = i >> 5;
            const int c4 = i & 31;
            const float4 v = ((const float4*)(h + (size_t)(row_base + r) * D))[c4];
            float* dp = &sA[r * SA_STRIDE + c4 * 4];
            dp[0] = v.x; dp[1] = v.y; dp[2] = v.z; dp[3] = v.w;
        }
    } else {
        for (int i = t; i < 16 * 32; i += 256) {
            const int r  = i >> 5;
            const int c4 = i & 31;
            const int row = row_base + r;
            float4 v = make_float4(0.f, 0.f, 0.f, 0.f);
            if (row < n_nodes)
                v = ((const float4*)(h + (size_t)row * D))[c4];
            float* dp = &sA[r * SA_STRIDE + c4 * 4];
            dp[0] = v.x; dp[1] = v.y; dp[2] = v.z; dp[3] = v.w;
        }
    }
    __syncthreads();

    const int lane = t & 31;        // wave32
    const int wv   = t >> 5;        // wave id 0..7 -> N slice
    const int half = lane >> 4;     // lane group (0: lanes 0-15, 1: 16-31)
    const int nloc = lane & 15;
    const int n    = wv * 16 + nloc;   // global output column
    const int ma   = nloc;             // A-matrix row for this lane

    v8f acc = {0.f, 0.f, 0.f, 0.f, 0.f, 0.f, 0.f, 0.f};

    #pragma unroll 4
    for (int kb = 0; kb < D; kb += 4) {
        const int k0 = kb + half * 2;   // per-lane K pair (ISA A/B layout)
        v2f a, bb;
        a.x  = sA[ma * SA_STRIDE + k0];
        a.y  = sA[ma * SA_STRIDE + k0 + 1];
        bb.x = W[(size_t)k0 * D + n];
        bb.y = W[(size_t)(k0 + 1) * D + n];
        acc = __builtin_amdgcn_wmma_f32_16x16x4_f32(
            /*neg_a=*/false, a, /*neg_b=*/false, bb,
            /*c_mod=*/(short)0, acc, /*reuse_a=*/false, /*reuse_b=*/false);
    }

    // D layout: VGPR i -> row (i + half*8), col = nloc within the wave slice.
    float* xp = x + (size_t)(row_base + half * 8) * D + n;
    if (full_tile) {
        #pragma unroll
        for (int i = 0; i < 8; ++i)
            xp[(size_t)i * D] = acc[i];
    } else {
        #pragma unroll
        for (int i = 0; i < 8; ++i) {
            if (row_base + half * 8 + i < n_nodes)
                xp[(size_t)i * D] = acc[i];
        }
    }
}

// ---------------------------------------------------------------------------
// Kernel 3: edge scatter — one wave per edge, float4 per lane (32*4 = 128).
// Edge index forced into an SGPR via readfirstlane so the metadata reads
// lower to scalar loads and the gather/atomics use saddr forms.
// agg[dst] += x[src] * w  via global_atomic_add_f32 (L2-resident operands).
// ---------------------------------------------------------------------------
__global__ __launch_bounds__(256)
void gcn_edge_scatter_kernel(const float* __restrict__ x,
                             const int* __restrict__ esrc,
                             const int* __restrict__ edst,
                             const float* __restrict__ ew,
                             float* __restrict__ agg,
                             int n_edges) {
    const int lane    = threadIdx.x & 31;
    const int wave_in = __builtin_amdgcn_readfirstlane(threadIdx.x >> 5);
    const long long e = (long long)blockIdx.x * 8 + wave_in;   // SGPR-uniform
    if (e >= n_edges) return;

    const int   s  = esrc[e];   // scalar load
    const int   d  = edst[e];   // scalar load
    const float wt = ew[e];     // scalar load

    const float4 v = ((const float4*)(x + (size_t)s * D))[lane];
    float* ad = agg + (size_t)d * D + lane * 4;
    atomicAdd(ad + 0, v.x * wt);
    atomicAdd(ad + 1, v.y * wt);
    atomicAdd(ad + 2, v.z * wt);
    atomicAdd(ad + 3, v.w * wt);
}

// ---------------------------------------------------------------------------
// Kernel 4: out = relu(agg + b), in place on d_out.
// ---------------------------------------------------------------------------
__global__ void gcn_bias_relu_kernel(float4* __restrict__ out,
                                     const float* __restrict__ b,
                                     long long n4) {
    long long i = (long long)blockIdx.x * blockDim.x + threadIdx.x;
    if (i >= n4) return;
    const int c = (int)((i * 4) & (D - 1));
    float4 v = out[i];
    v.x = fmaxf(v.x + b[c + 0], 0.f);
    v.y = fmaxf(v.y + b[c + 1], 0.f);
    v.z = fmaxf(v.z + b[c + 2], 0.f);
    v.w = fmaxf(v.w + b[c + 3], 0.f);
    out[i] = v;
}

// ---------------------------------------------------------------------------
extern "C" void kernel_launch(void* const* d_in, const int* in_sizes, int n_in,
                              void* d_out, int out_size, void* d_ws, size_t ws_size,
                              hipStream_t stream) {
    const float* h    = (const float*)d_in[0];
    const int*   esrc = (const int*)d_in[1];
    const int*   edst = (const int*)d_in[2];
    const float* ew   = (const float*)d_in[3];
    const float* W    = (const float*)d_in[4];
    const float* b    = (const float*)d_in[5];

    const int n_nodes = in_sizes[0] / D;
    const int n_edges = in_sizes[1];

    float* out = (float*)d_out;          // doubles as agg accumulator
    float* x   = (float*)d_ws;           // n_nodes * 128 floats of scratch

    const long long n4 = (long long)n_nodes * (D / 4);

    // 1) agg = 0
    {
        const int blk = 256;
        const long long grid = (n4 + blk - 1) / blk;
        gcn_zero_kernel<<<(unsigned)grid, blk, 0, stream>>>((float4*)out, n4);
    }
    // 2) x = h @ W (WMMA fp32)
    {
        const int grid = (n_nodes + 15) / 16;
        gcn_gemm_wmma_kernel<<<grid, 256, 0, stream>>>(h, W, x, n_nodes);
    }
    // 3) agg[dst] += x[src] * w   (one wave per edge, 8 waves per block)
    {
        const int blk = 256;
        const int grid = (n_edges + 7) / 8;
        gcn_edge_scatter_kernel<<<grid, blk, 0, stream>>>(
            x, esrc, edst, ew, out, n_edges);
    }
    // 4) out = relu(agg + b)
    {
        const int blk = 256;
        const long long grid = (n4 + blk - 1) / blk;
        gcn_bias_relu_kernel<<<(unsigned)grid, blk, 0, stream>>>(
            (float4*)out, b, n4);
    }
}